// MultiInSizeLinear_20306605375707
// MI455X (gfx1250) — compile-verified
//
#include <hip/hip_runtime.h>

typedef __attribute__((ext_vector_type(2))) float v2f;
typedef __attribute__((ext_vector_type(8))) float v8f;

#define N_TOK   32768      // 16 * 2048 tokens
#define MAXF    128
#define OUTF    512
#define NEXP    5
#define ROW_PAD 132        // 128 + 4 floats: bank-conflict-free, rows 16B aligned

__global__ __launch_bounds__(128)
void multi_in_linear_wmma(const float* __restrict__ x,
                          const int*   __restrict__ feat_sz,
                          const float* __restrict__ weight,
                          const float* __restrict__ bias,
                          float* __restrict__ out)
{
    __shared__ __align__(16) float xs[16 * ROW_PAD];
    __shared__ int exs[16];

    const int tile = blockIdx.x;           // 16-token tile
    const int tid  = threadIdx.x;
    const int lane = tid & 31;
    const int wave = tid >> 5;             // 4 wave32 per block

    // ---- stage x tile (16 x 128 f32) into LDS, coalesced float4 ----
    const float* xt = x + (size_t)tile * 16 * MAXF;
    for (int i = tid; i < 16 * (MAXF / 4); i += 128) {
        const int row = i >> 5;            // 32 float4 per row
        const int c4  = i & 31;
        const float4 v = reinterpret_cast<const float4*>(xt + row * MAXF)[c4];
        reinterpret_cast<float4*>(&xs[row * ROW_PAD])[c4] = v;
    }
    if (tid < 16) {
        const int sz = feat_sz[tile * 16 + tid];
        int e = 4;                         // size -> expert index in {8,16,32,64,128}
        if      (sz == 8)  e = 0;
        else if (sz == 16) e = 1;
        else if (sz == 32) e = 2;
        else if (sz == 64) e = 3;
        exs[tid] = e;
    }
    __syncthreads();

    const int nl   = lane & 15;            // N within 16-wide out tile
    const int hi   = lane >> 4;            // half-wave selector
    const int koff = hi * 2;               // K sub-offset for A/B fragments
    const int tokA = nl;                   // A-matrix row (token) for this lane

    // expert index of the 8 tokens this lane's C/D registers map to:
    // lane L, vgpr r -> M = r + 8*(L>=16)
    int etok[8];
    #pragma unroll
    for (int r = 0; r < 8; ++r) etok[r] = exs[r + 8 * hi];

    // each wave covers out-tiles wave, wave+4, ... -> 8 tiles of 16 = 128 outs per wave
    for (int ot = wave; ot < OUTF / 16; ot += 4) {
        const int o = ot * 16 + nl;
        v8f fac = {};

        #pragma unroll
        for (int e = 0; e < NEXP; ++e) {
            const int size = 8 << e;
            const float* wrow = weight + ((size_t)(e * OUTF + o)) * MAXF + koff;
            v8f acc = {};
            #pragma unroll 8
            for (int k = 0; k < size; k += 4) {
                // A 16x4 f32: lane<16 holds K={0,1}, lane>=16 holds K={2,3} of row M=lane%16
                v2f a = *reinterpret_cast<const v2f*>(&xs[tokA * ROW_PAD + k + koff]);
                // B 4x16 f32: vgpr0 = rows K={0,2}, vgpr1 = rows K={1,3}, N = lane%16
                v2f b = *reinterpret_cast<const v2f*>(wrow + k);
                acc = __builtin_amdgcn_wmma_f32_16x16x4_f32(
                        false, a, false, b, (short)0, acc, false, false);
            }
            const float bv = bias[e * OUTF + o];
            #pragma unroll
            for (int r = 0; r < 8; ++r) {
                const float sel = (etok[r] == e) ? 1.0f : 0.0f;
                fac[r] += sel * (acc[r] + bv);   // one-hot select, faithful to reference
            }
        }

        // C/D layout: lane L, vgpr r -> M = r + 8*hi, N = nl ; coalesced f32 stores
        float* orow = out + (size_t)tile * 16 * OUTF + ot * 16 + nl;
        #pragma unroll
        for (int r = 0; r < 8; ++r) {
            orow[(size_t)(r + 8 * hi) * OUTF] = fac[r];
        }
    }
}

extern "C" void kernel_launch(void* const* d_in, const int* in_sizes, int n_in,
                              void* d_out, int out_size, void* d_ws, size_t ws_size,
                              hipStream_t stream)
{
    const float* x  = (const float*)d_in[0];   // [16,2048,128] f32
    const int*   fs = (const int*)d_in[1];     // [16,2048] int32 sizes
    const float* w  = (const float*)d_in[2];   // [5,512,128] f32
    const float* b  = (const float*)d_in[3];   // [5,512] f32
    float* out = (float*)d_out;                // [16,2048,512] f32

    dim3 grid(N_TOK / 16);    // 2048 token tiles
    dim3 block(128);          // 4 wave32
    multi_in_linear_wmma<<<grid, block, 0, stream>>>(x, fs, w, b, out);
}